// SecDecoder_17798344475040
// MI455X (gfx1250) — compile-verified
//
#include <hip/hip_runtime.h>
#include <math.h>

typedef __attribute__((ext_vector_type(2))) float v2f;
typedef __attribute__((ext_vector_type(4))) float v4f;
typedef __attribute__((ext_vector_type(8))) float v8f;

#define NCOL 1024
#define DCOL 512
#define NOUT 26

// ---------------------------------------------------------------------------
// Kernel 1: soft_argmax row reduction (memory-bound streaming kernel).
// One wave32 per row of M; each lane does 8x float4 non-temporal loads
// (coalesced 512B per instruction across the wave), weights by the column
// index (computed, not loaded), then a 5-step shuffle reduction.
// rel[i] = (M[i,:] . arange(NCOL)) / NCOL   (division by 2^10 is exact)
// ---------------------------------------------------------------------------
__global__ __launch_bounds__(256) void rowdot_kernel(const float* __restrict__ M,
                                                     float* __restrict__ rel,
                                                     int total) {
  int wid = blockIdx.x * 8 + (threadIdx.x >> 5);
  int lane = threadIdx.x & 31;
  if (wid >= total) return;
  const float* row = M + (size_t)wid * NCOL;
  float acc = 0.0f;
#pragma unroll
  for (int u = 0; u < 8; ++u) {
    int c = u * 128 + lane * 4;  // contiguous 128-col stripe per unroll step
    v4f v = __builtin_nontemporal_load((const v4f*)(row + c));
    float fc = (float)c;
    acc = fmaf(v.x, fc,        acc);
    acc = fmaf(v.y, fc + 1.0f, acc);
    acc = fmaf(v.z, fc + 2.0f, acc);
    acc = fmaf(v.w, fc + 3.0f, acc);
  }
#pragma unroll
  for (int off = 16; off > 0; off >>= 1) acc += __shfl_down(acc, off, 32);
  if (lane == 0) rel[wid] = acc * (1.0f / (float)NCOL);
}

// ---------------------------------------------------------------------------
// Kernel 2: tiny serial cumsum of sequence lengths (500 entries).
// cs[k] = sum_{i<=k} len[i]
// ---------------------------------------------------------------------------
__global__ void cumsum_kernel(const int* __restrict__ len, int* __restrict__ cs,
                              int nseq) {
  if (threadIdx.x == 0 && blockIdx.x == 0) {
    int acc = 0;
    for (int i = 0; i < nseq; ++i) { acc += len[i]; cs[i] = acc; }
  }
}

// ---------------------------------------------------------------------------
// Kernel 3: gap outputs. One block per sequence k with rows [Bk, Ek).
//   gaps_start[k] = rel[Bk]
//   gaps_end[k]   = (NCOL-1)/NCOL - rel[Ek-1]
//   internal gap g in [Bk, Ek-1): gin[g - k] = rel[g+1] - rel[g] - 1/NCOL
// (all bit-identical to reference because /2^10 commutes with rounding)
// ---------------------------------------------------------------------------
__global__ __launch_bounds__(256) void gaps_kernel(const float* __restrict__ rel,
                                                   const int* __restrict__ cs,
                                                   float* __restrict__ gstart,
                                                   float* __restrict__ gin,
                                                   float* __restrict__ gend) {
  int k = blockIdx.x;
  int Bk = (k == 0) ? 0 : cs[k - 1];
  int Ek = cs[k];
  if (threadIdx.x == 0) {
    gstart[k] = rel[Bk];
    gend[k] = ((float)(NCOL - 1) / (float)NCOL) - rel[Ek - 1];
  }
  int outbase = Bk - k;  // sum_{m<k} (len_m - 1)
  for (int g = Bk + (int)threadIdx.x; g < Ek - 1; g += (int)blockDim.x) {
    gin[outbase + (g - Bk)] = rel[g + 1] - rel[g] - (1.0f / (float)NCOL);
  }
}

// ---------------------------------------------------------------------------
// Kernel 4: col_dist = softmax(columns @ W + b) via V_WMMA_F32_16X16X4_F32.
// One wave32 per 16-row tile of `columns`; two 16-wide N tiles (N 0..31)
// accumulated over K=512 in steps of 4 -> 256 WMMAs/wave.
//
// Fragment layouts (CDNA5 ISA 7.12.2):
//   A (16x4 f32): lanes 0-15 -> M=lane, v[0]=K0,v[1]=K1; lanes 16-31 -> K2,K3
//   B ( 4x16 f32): lane&15 -> N;   lanes 0-15 -> v[0]=K0,v[1]=K1; hi -> K2,K3
//   C/D (16x16 f32): v[r] -> M=r (lanes 0-15) / M=r+8 (lanes 16-31), N=lane&15
//
// N>=26 lanes of the second tile load a *clamped* (in-bounds) W address
// instead of a predicated zero: output element (m,n) depends only on B
// column n, and columns n>=26 are never stored, so their values are
// irrelevant. This keeps the hot loop free of EXEC-mask branches so the
// WMMAs issue back-to-back with plain coalesced loads.
// ---------------------------------------------------------------------------
__global__ __launch_bounds__(32) void coldist_kernel(const float* __restrict__ cols,
                                                     const float* __restrict__ W,
                                                     const float* __restrict__ bias,
                                                     float* __restrict__ out) {
  __shared__ float tile[16 * 32];
  const int lane = threadIdx.x;       // 0..31, EXEC all-ones (no divergence)
  const int grp  = lane >> 4;         // lane half: selects K offset
  const int l    = lane & 15;
  const int rb   = blockIdx.x * 16;   // row base in columns
  const int kofs = grp * 2;
  const int n1c  = (16 + l < NOUT) ? (16 + l) : (NOUT - 1);  // clamped, hoisted

  const float* arow = cols + (size_t)(rb + l) * DCOL;

  v8f c0 = {};  // N = 0..15
  v8f c1 = {};  // N = 16..31 (only 16..25 valid / stored)
  for (int k = 0; k < DCOL; k += 4) {
    v2f a;
    a.x = arow[k + kofs];
    a.y = arow[k + kofs + 1];

    const float* w0 = W + (size_t)(k + kofs) * NOUT;      // row K=k+kofs
    const float* w1 = w0 + NOUT;                          // row K=k+kofs+1
    v2f b0, b1;
    b0.x = w0[l];
    b0.y = w1[l];
    b1.x = w0[n1c];
    b1.y = w1[n1c];

    c0 = __builtin_amdgcn_wmma_f32_16x16x4_f32(false, a, false, b0,
                                               (short)0, c0, false, false);
    c1 = __builtin_amdgcn_wmma_f32_16x16x4_f32(false, a, false, b1,
                                               (short)0, c1, false, false);
  }

  // Scatter C fragments to LDS as [16 rows][32 cols] for the softmax pass.
#pragma unroll
  for (int r = 0; r < 8; ++r) {
    int row = grp * 8 + r;
    tile[row * 32 + l]      = c0[r];
    tile[row * 32 + 16 + l] = c1[r];
  }
  __syncthreads();

  // Softmax over the 26 classes: one lane per row (lanes 0..15).
  if (lane < 16) {
    int row = lane;
    float logit[NOUT];
    float mx = -1e30f;
#pragma unroll
    for (int n = 0; n < NOUT; ++n) {
      logit[n] = tile[row * 32 + n] + bias[n];
      mx = fmaxf(mx, logit[n]);
    }
    float s = 0.0f;
#pragma unroll
    for (int n = 0; n < NOUT; ++n) {
      logit[n] = __expf(logit[n] - mx);
      s += logit[n];
    }
    float inv = 1.0f / s;
    float* o = out + (size_t)(rb + row) * NOUT;
#pragma unroll
    for (int n = 0; n < NOUT; ++n) o[n] = logit[n] * inv;
  }
}

// ---------------------------------------------------------------------------
// Launch wrapper.
// d_out layout (flat, reference return order):
//   [0, total)                       relative_positions
//   [total, total+nseq)              gaps_start
//   [total+nseq, 2*total)            gaps_in      (total - nseq entries)
//   [2*total, 2*total+nseq)          gaps_end
//   [2*total+nseq, ...)              col_dist     (NCOL*NOUT)
// ---------------------------------------------------------------------------
extern "C" void kernel_launch(void* const* d_in, const int* in_sizes, int n_in,
                              void* d_out, int out_size, void* d_ws, size_t ws_size,
                              hipStream_t stream) {
  const float* M       = (const float*)d_in[0];
  const float* columns = (const float*)d_in[1];
  const int*   seqlen  = (const int*)d_in[2];
  const float* W       = (const float*)d_in[3];
  const float* bias    = (const float*)d_in[4];
  float* out = (float*)d_out;

  int total = in_sizes[0] / NCOL;   // 100000
  int nseq  = in_sizes[2];          // 500

  float* rel     = out;                    // [total]
  float* gstart  = out + total;            // [nseq]
  float* gin     = gstart + nseq;          // [total - nseq]
  float* gend    = gin + (total - nseq);   // [nseq]
  float* coldist = gend + nseq;            // [NCOL * NOUT]

  int* cs = (int*)d_ws;                    // [nseq] cumsum scratch

  rowdot_kernel<<<(total + 7) / 8, 256, 0, stream>>>(M, rel, total);
  cumsum_kernel<<<1, 32, 0, stream>>>(seqlen, cs, nseq);
  gaps_kernel<<<nseq, 256, 0, stream>>>(rel, cs, gstart, gin, gend);
  coldist_kernel<<<NCOL / 16, 32, 0, stream>>>(columns, W, bias, coldist);
}